// GCN_3066606649548
// MI455X (gfx1250) — compile-verified
//
#include <hip/hip_runtime.h>
#include <math.h>

// GCN forward on MI455X (gfx1250, wave32).
// Dense GEMMs use V_WMMA_F32_16X16X4_F32 (exact fp32) with A staged via LDS
// and B pre-packed into fragment order (1 coalesced global_load_b64 per WMMA,
// immediate-offset addressing). SpMM uses hardware f32 global atomics
// (accumulators are 25.6MB << 192MB L2, so scatter traffic stays on-chip).

typedef __attribute__((ext_vector_type(2))) float v2f;
typedef __attribute__((ext_vector_type(8))) float v8f;

#define NFEAT   512
#define NHID    128
#define NCLASS  40
#define KCHUNK  64
#define LDSTR   68   // padded LDS row stride (floats): conflict-free across M-lanes

// ---------------------------------------------------------------------------
// Pack B[K][128] (row-major) into WMMA fragment order:
// element i = (ntile*(K/4) + f)*32 + lane holds the v2f
//   { B[f*4 + 2*(lane>>4)][ntile*16 + (lane&15)],
//     B[f*4 + 2*(lane>>4) + 1][ntile*16 + (lane&15)] }
// so in the GEMM each lane loads one contiguous 8B element per k-step, and
// consecutive fragments are 256B apart (immediate offsets, perfect coalescing).
// ---------------------------------------------------------------------------
__global__ void pack_B(const float* __restrict__ B, float* __restrict__ Bp, int K)
{
  const int frag_per_tile = K >> 2;
  const int total = 8 * frag_per_tile * 32;
  int i = blockIdx.x * blockDim.x + threadIdx.x;
  if (i >= total) return;
  const int lane  = i & 31;
  const int f     = (i >> 5) % frag_per_tile;
  const int ntile = (i >> 5) / frag_per_tile;
  const int n = (ntile << 4) + (lane & 15);
  const int k = (f << 2) + ((lane >> 4) << 1);
  float2 v = make_float2(B[(size_t)k * NHID + n], B[(size_t)(k + 1) * NHID + n]);
  *reinterpret_cast<float2*>(Bp + (size_t)i * 2) = v;
}

// ---------------------------------------------------------------------------
// C[M,128] = A[M,K] @ B[K,128]; one block = 16 rows x full N (8 waves).
// A tile staged through LDS in 16x64 chunks; wave w owns N columns [16w,16w+16).
// ---------------------------------------------------------------------------
template<int K>
__global__ __launch_bounds__(256) void gemm_wmma_f32(
    const float* __restrict__ A, const float* __restrict__ Bpack,
    float* __restrict__ C, int M)
{
  __shared__ float As[16 * LDSTR];

  const int tid  = threadIdx.x;
  const int lane = tid & 31;
  const int wave = tid >> 5;         // 0..7 -> N tile
  const int l16  = lane & 15;
  const int half = lane >> 4;
  const int m0   = blockIdx.x << 4;

  // cooperative A-chunk load: thread t -> row t/16, 4 cols at (t%16)*4
  const int arow = tid >> 4;
  const int acol = (tid & 15) << 2;
  const int asrc = (m0 + arow < M) ? (m0 + arow) : (M - 1);
  const float* Aload = A + (size_t)asrc * K + acol;
  float* AsStore = &As[arow * LDSTR + acol];

  // this lane's fragment streams (constant-stride -> immediate offsets)
  const float* bp = Bpack + ((size_t)wave * (K >> 2) * 32 + lane) * 2;
  const float* arow_lds = &As[l16 * LDSTR + (half << 1)];

  v8f acc = {};
  for (int kc = 0; kc < K; kc += KCHUNK) {
    float4 av = *reinterpret_cast<const float4*>(Aload + kc);
    __syncthreads();                                   // previous chunk consumed
    *reinterpret_cast<float4*>(AsStore) = av;
    __syncthreads();
#pragma unroll
    for (int k0 = 0; k0 < KCHUNK; k0 += 4) {
      v2f a = *reinterpret_cast<const v2f*>(arow_lds + k0);
      v2f b = *reinterpret_cast<const v2f*>(bp + (k0 >> 2) * 64);
      acc = __builtin_amdgcn_wmma_f32_16x16x4_f32(false, a, false, b,
                                                  (short)0, acc, false, false);
    }
    bp += (KCHUNK >> 2) * 64;
  }

  // C/D layout: VGPR r = row (r + 8*half), col = l16 within wave's N tile
  const int crow = m0 + (half << 3);
  float* Crow = C + (size_t)crow * NHID + (wave << 4) + l16;
#pragma unroll
  for (int r = 0; r < 8; ++r) {
    if (crow + r < M) Crow[(size_t)r * NHID] = acc[r];
  }
}

// ---------------------------------------------------------------------------
__global__ void zero_f32(float* __restrict__ p, int n4)
{
  int i = blockIdx.x * blockDim.x + threadIdx.x;
  if (i < n4) reinterpret_cast<float4*>(p)[i] = make_float4(0.f, 0.f, 0.f, 0.f);
}

// One wave per edge: gather source row (float4/lane), scale, atomic scatter-add.
__global__ __launch_bounds__(256) void spmm_scatter(
    const int* __restrict__ rows, const int* __restrict__ cols,
    const float* __restrict__ vals, const float* __restrict__ dense,
    float* __restrict__ out, int E)
{
  const int lane = threadIdx.x & 31;
  const int e = blockIdx.x * 8 + (threadIdx.x >> 5);
  if (e >= E) return;
  const int   r = rows[e];
  const int   c = cols[e];
  const float v = vals[e];
  float4 s = reinterpret_cast<const float4*>(dense + (size_t)c * NHID)[lane];
  float* dst = out + (size_t)r * NHID + (lane << 2);
  unsafeAtomicAdd(dst + 0, s.x * v);
  unsafeAtomicAdd(dst + 1, s.y * v);
  unsafeAtomicAdd(dst + 2, s.z * v);
  unsafeAtomicAdd(dst + 3, s.w * v);
}

__global__ void bias_relu(float* __restrict__ h, const float* __restrict__ bias,
                          int total)
{
  int i = blockIdx.x * blockDim.x + threadIdx.x;
  if (i >= total) return;
  float x = h[i] + bias[i & (NHID - 1)];
  h[i] = x > 0.f ? x : 0.f;
}

// One wave per node row: logits = h@Wout + bout (lane -> class lane, lane+32),
// then log_softmax over 40 classes via wave32 shuffle reductions.
__global__ __launch_bounds__(256) void out_logsoftmax(
    const float* __restrict__ H, const float* __restrict__ Wout,
    const float* __restrict__ bout, float* __restrict__ out, int M)
{
  const int lane = threadIdx.x & 31;
  const int row  = blockIdx.x * 8 + (threadIdx.x >> 5);
  if (row >= M) return;

  const float* h = H + (size_t)row * NHID;
  const int c0 = lane;
  const int c1 = lane + 32;
  const bool has2 = (c1 < NCLASS);

  float acc0 = bout[c0];
  float acc1 = has2 ? bout[c1] : 0.f;
  for (int k = 0; k < NHID; ++k) {
    float hk = h[k];
    const float* wr = Wout + k * NCLASS;
    acc0 = fmaf(hk, wr[c0], acc0);
    if (has2) acc1 = fmaf(hk, wr[c1], acc1);
  }

  float l1 = has2 ? acc1 : -__builtin_inff();
  float m = fmaxf(acc0, l1);
#pragma unroll
  for (int off = 16; off > 0; off >>= 1)
    m = fmaxf(m, __shfl_xor(m, off, 32));
  float s = expf(acc0 - m) + (has2 ? expf(l1 - m) : 0.f);
#pragma unroll
  for (int off = 16; off > 0; off >>= 1)
    s += __shfl_xor(s, off, 32);
  float lse = m + logf(s);

  out[(size_t)row * NCLASS + c0] = acc0 - lse;
  if (has2) out[(size_t)row * NCLASS + c1] = l1 - lse;
}

// ---------------------------------------------------------------------------
extern "C" void kernel_launch(void* const* d_in, const int* in_sizes, int n_in,
                              void* d_out, int out_size, void* d_ws, size_t ws_size,
                              hipStream_t stream)
{
  const float* x    = (const float*)d_in[0];
  const int*   rows = (const int*)  d_in[1];
  const int*   cols = (const int*)  d_in[2];
  const float* vals = (const float*)d_in[3];
  const float* W1   = (const float*)d_in[4];
  const float* b1   = (const float*)d_in[5];
  const float* W2   = (const float*)d_in[6];
  const float* b2   = (const float*)d_in[7];
  const float* Wout = (const float*)d_in[8];
  const float* bout = (const float*)d_in[9];
  float* out = (float*)d_out;

  const int M = in_sizes[0] / NFEAT;   // 50000 nodes
  const int E = in_sizes[1];           // 1.6M edges

  float* support = (float*)d_ws;                        // [M, NHID]
  float* accum   = support + (size_t)M * NHID;          // [M, NHID]
  float* W1p     = accum   + (size_t)M * NHID;          // NFEAT*NHID packed
  float* W2p     = W1p     + (size_t)NFEAT * NHID;      // NHID*NHID packed

  const int total  = M * NHID;
  const int gM     = (M + 15) / 16;
  const int gZero  = (total / 4 + 255) / 256;
  const int gElem  = (total + 255) / 256;
  const int gEdge  = (E + 7) / 8;
  const int gRow   = (M + 7) / 8;

  // Pack weights into WMMA fragment order (cheap: 64K + 16K elements).
  pack_B<<<(8 * (NFEAT / 4) * 32 + 255) / 256, 256, 0, stream>>>(W1, W1p, NFEAT);
  pack_B<<<(8 * (NHID  / 4) * 32 + 255) / 256, 256, 0, stream>>>(W2, W2p, NHID);

  // Layer 1: support = x @ W1 ; h1 = relu(spmm(adj, support) + b1)
  gemm_wmma_f32<NFEAT><<<gM, 256, 0, stream>>>(x, W1p, support, M);
  zero_f32<<<gZero, 256, 0, stream>>>(accum, total / 4);
  spmm_scatter<<<gEdge, 256, 0, stream>>>(rows, cols, vals, support, accum, E);
  bias_relu<<<gElem, 256, 0, stream>>>(accum, b1, total);

  // Layer 2: support = h1 @ W2 ; h2 = relu(spmm(adj, support) + b2)
  gemm_wmma_f32<NHID><<<gM, 256, 0, stream>>>(accum, W2p, support, M);
  zero_f32<<<gZero, 256, 0, stream>>>(accum, total / 4);
  spmm_scatter<<<gEdge, 256, 0, stream>>>(rows, cols, vals, support, accum, E);
  bias_relu<<<gElem, 256, 0, stream>>>(accum, b2, total);

  // Output: log_softmax(h2 @ Wout + bout)
  out_logsoftmax<<<gRow, 256, 0, stream>>>(accum, Wout, bout, out, M);
}